// Quantizer_12584254177997
// MI455X (gfx1250) — compile-verified
//
#include <hip/hip_runtime.h>

#define QNBINS 128

// ---- total-order float <-> uint mapping (exact min/max via integer atomics) ----
__device__ __forceinline__ unsigned ordf(float f) {
  unsigned u = __float_as_uint(f);
  return (u & 0x80000000u) ? ~u : (u | 0x80000000u);
}
__device__ __forceinline__ float deordf(unsigned v) {
  unsigned u = (v & 0x80000000u) ? (v ^ 0x80000000u) : ~v;
  return __uint_as_float(u);
}

// ws layout: ws[0..7] = ordered column mins, ws[8..15] = ordered column maxs
__global__ void init_minmax(unsigned* __restrict__ ws) {
  int t = threadIdx.x;
  if (t < 8)        ws[t] = 0xFFFFFFFFu;  // identity for ordered-min (= +inf)
  else if (t < 16)  ws[t] = 0x00000000u;  // identity for ordered-max (= -inf)
}

// Each lane loads float4 (B128). Row stride is 8 floats, so a lane's float4
// covers columns (parity*4 .. parity*4+3) with parity = tid&1, constant across
// the grid-stride loop (stride is even). Butterfly over xor offsets 2,4,8,16
// reduces within the parity class; lanes 0 and 1 publish via integer atomics.
__global__ void colwise_minmax(const float4* __restrict__ f4, int n4,
                               unsigned* __restrict__ ws) {
  int tid    = blockIdx.x * blockDim.x + threadIdx.x;
  int stride = gridDim.x * blockDim.x;          // even by construction
  float mn0 =  INFINITY, mn1 =  INFINITY, mn2 =  INFINITY, mn3 =  INFINITY;
  float mx0 = -INFINITY, mx1 = -INFINITY, mx2 = -INFINITY, mx3 = -INFINITY;
  for (int i = tid; i < n4; i += stride) {
    float4 v = f4[i];
    mn0 = fminf(mn0, v.x); mx0 = fmaxf(mx0, v.x);
    mn1 = fminf(mn1, v.y); mx1 = fmaxf(mx1, v.y);
    mn2 = fminf(mn2, v.z); mx2 = fmaxf(mx2, v.z);
    mn3 = fminf(mn3, v.w); mx3 = fmaxf(mx3, v.w);
  }
  // wave32 butterfly, parity-preserving offsets only
  #pragma unroll
  for (int off = 2; off < 32; off <<= 1) {
    mn0 = fminf(mn0, __shfl_xor(mn0, off, 32));
    mn1 = fminf(mn1, __shfl_xor(mn1, off, 32));
    mn2 = fminf(mn2, __shfl_xor(mn2, off, 32));
    mn3 = fminf(mn3, __shfl_xor(mn3, off, 32));
    mx0 = fmaxf(mx0, __shfl_xor(mx0, off, 32));
    mx1 = fmaxf(mx1, __shfl_xor(mx1, off, 32));
    mx2 = fmaxf(mx2, __shfl_xor(mx2, off, 32));
    mx3 = fmaxf(mx3, __shfl_xor(mx3, off, 32));
  }
  int lane = threadIdx.x & 31;
  if (lane < 2) {
    int base = lane * 4;  // lane0 -> cols 0..3, lane1 -> cols 4..7
    atomicMin(&ws[base + 0], ordf(mn0));
    atomicMin(&ws[base + 1], ordf(mn1));
    atomicMin(&ws[base + 2], ordf(mn2));
    atomicMin(&ws[base + 3], ordf(mn3));
    atomicMax(&ws[8 + base + 0], ordf(mx0));
    atomicMax(&ws[8 + base + 1], ordf(mx1));
    atomicMax(&ws[8 + base + 2], ordf(mx2));
    atomicMax(&ws[8 + base + 3], ordf(mx3));
  }
}

// exact reference arithmetic (no FMA contraction, half-to-even rounding)
__device__ __forceinline__ float lin_at(float fmin, float d, int b) {
  float t = __fdiv_rn((float)b, 127.0f);          // arange(128)/127 in f32
  return __fadd_rn(fmin, __fmul_rn(d, t));        // fmin + d*t, separate rounds
}
// sign of round6(lin[b]-feat): rintf output is integer-valued and /1e6 with a
// positive constant preserves sign, so comparisons can skip the division.
__device__ __forceinline__ float diffr(float fmin, float d, int b, float feat) {
  return rintf(__fmul_rn(__fsub_rn(lin_at(fmin, d, b), feat), 1.0e6f));
}

// 4 consecutive elements per thread: element e = 4*i + j, var = (i&1)*4 + j.
// B128 async stage into LDS, B128 stores for both outputs.
__global__ void bin_kernel4(const float4* __restrict__ feats4,
                            const unsigned* __restrict__ ws,
                            float4* __restrict__ out_bins,
                            float4* __restrict__ out_regs, int n4) {
  __shared__ float4 tile[256];
  int i = blockIdx.x * blockDim.x + threadIdx.x;
  if (i < n4) {
    // CDNA5 async global->LDS staging (ASYNCcnt-tracked). GVS mode:
    // each lane copies 16 bytes from the base in saddr plus the byte offset in vaddr to the LDS address in vdst.
    unsigned ldsA = (unsigned)(size_t)&tile[threadIdx.x];
    unsigned voff = (unsigned)i * 16u;
    asm volatile("global_load_async_to_lds_b128 %0, %1, %2\n\t"
                 "s_wait_asynccnt 0"
                 :
                 : "v"(ldsA), "v"(voff), "s"(feats4)
                 : "memory");
    float4 f = tile[threadIdx.x];

    int  cbase = (i & 1) * 4;               // this thread's column block
    float fe[4] = {f.x, f.y, f.z, f.w};
    float bo[4], ro[4];
    #pragma unroll
    for (int j = 0; j < 4; ++j) {
      int   var  = cbase + j;
      float fmn  = deordf(ws[var]);
      float fmx  = deordf(ws[8 + var]);
      float d    = __fsub_rn(fmx, fmn);
      // step = lin[1] - lin[0] exactly as the reference computes it
      float step = __fsub_rn(lin_at(fmn, d, 1), fmn);
      float feat = fe[j];

      // floor estimate, then exact correction under the 1e-6 rounded comparison
      int b = 0;
      if (step > 0.0f) {
        float q = __fdiv_rn(__fsub_rn(feat, fmn), step);
        q = fminf(fmaxf(q, 0.0f), 127.0f);
        b = (int)floorf(q);
      }
      while (b < QNBINS - 1 && diffr(fmn, d, b + 1, feat) <= 0.0f) ++b;
      while (b > 0          && diffr(fmn, d, b,     feat) >  0.0f) --b;

      float val = lin_at(fmn, d, b);
      bo[j] = (float)b;
      ro[j] = __fdiv_rn(fmaxf(__fsub_rn(feat, val), 0.0f), step);
    }
    out_bins[i] = make_float4(bo[0], bo[1], bo[2], bo[3]);
    out_regs[i] = make_float4(ro[0], ro[1], ro[2], ro[3]);
  }
}

extern "C" void kernel_launch(void* const* d_in, const int* in_sizes, int n_in,
                              void* d_out, int out_size, void* d_ws, size_t ws_size,
                              hipStream_t stream) {
  const float* feats = (const float*)d_in[0];
  int total = in_sizes[0];              // 100000 * 8 = 800000
  unsigned* ws = (unsigned*)d_ws;       // 16 x u32 used
  float* out = (float*)d_out;
  int half = out_size / 2;              // bins | regs, concatenated flat

  init_minmax<<<1, 32, 0, stream>>>(ws);

  int n4 = total / 4;                   // 8 cols -> always divisible by 4
  colwise_minmax<<<256, 256, 0, stream>>>((const float4*)feats, n4, ws);

  int bblocks = (n4 + 255) / 256;
  bin_kernel4<<<bblocks, 256, 0, stream>>>((const float4*)feats, ws,
                                           (float4*)out, (float4*)(out + half), n4);
}